// not_efficientMLP_15247133901512
// MI455X (gfx1250) — compile-verified
//
#include <hip/hip_runtime.h>

typedef float v2f __attribute__((ext_vector_type(2)));
typedef float v4f __attribute__((ext_vector_type(4)));
typedef float v8f __attribute__((ext_vector_type(8)));
typedef int   v4i __attribute__((ext_vector_type(4)));

typedef __attribute__((address_space(1))) v4i gv4i;  // global int4
typedef __attribute__((address_space(3))) v4i lv4i;  // LDS int4

#define N_ 512
#define D_ 1024
#define H_ 512

#if defined(__gfx1250__) && __has_builtin(__builtin_amdgcn_global_load_async_to_lds_b128)
#define USE_ASYNC_LDS 1
#else
#define USE_ASYNC_LDS 0
#endif

__device__ __forceinline__ void wait_async_lds() {
#if __has_builtin(__builtin_amdgcn_s_wait_asynccnt)
  __builtin_amdgcn_s_wait_asynccnt(0);
#else
  asm volatile("s_wait_asynccnt 0x0" ::: "memory");
#endif
}

// ---------------------------------------------------------------------------
// Kernel 1: A = X @ Wa^T + b1 (bias folded),  B = X @ Wb^T
// via V_WMMA_F32_16X16X4_F32 (exact f32 numerics).
// One wave per 16x16 (i,h) tile; X fragment reused for both WMMAs.
//
// f32 A-frag (16x4): lanes 0-15 -> M=lane, VGPRs {K=0,K=1}; lanes 16-31 ->
// M=lane-16, VGPRs {K=2,K=3}. B-frag (4x16) mirrors with N=lane&15. Both are
// a contiguous per-lane float2 at row = base+(lane&15), col = d0+2*(lane>>4).
// C/D: VGPR v -> M = v + 8*(lane>>4), N = h0 + (lane&15)  => b1[h0+(lane&15)]
// is a single per-lane scalar added to all 8 accumulator components.
// ---------------------------------------------------------------------------
__global__ void __launch_bounds__(256)
gemm_ab_kernel(const float* __restrict__ X, const float* __restrict__ W1,
               const float* __restrict__ b1,
               float* __restrict__ Amat, float* __restrict__ Bmat)
{
  const int lane = threadIdx.x & 31;
  const int wid  = (blockIdx.x << 3) | (threadIdx.x >> 5);   // 0..1023
  const int i0   = (wid >> 5) << 4;                          // i-tile base
  const int h0   = (wid & 31) << 4;                          // h-tile base
  const int r    = lane & 15;
  const int kq   = lane >> 4;                                // 0/1 -> K pair

  const float* xrow  = X  + (i0 + r) * D_       + 2 * kq;
  const float* warow = W1 + (h0 + r) * (2 * D_) + 2 * kq;    // Wa = W1[:, :D]
  const float* wbrow = warow + D_;                           // Wb = W1[:, D:]

  v8f accA = {0.f, 0.f, 0.f, 0.f, 0.f, 0.f, 0.f, 0.f};
  v8f accB = {0.f, 0.f, 0.f, 0.f, 0.f, 0.f, 0.f, 0.f};

  #pragma unroll 4
  for (int d0 = 0; d0 < D_; d0 += 4) {
    v2f xf = *reinterpret_cast<const v2f*>(xrow  + d0);
    v2f wa = *reinterpret_cast<const v2f*>(warow + d0);
    v2f wb = *reinterpret_cast<const v2f*>(wbrow + d0);
    // (neg_a, A, neg_b, B, c_mod, C, reuse_a, reuse_b)
    accA = __builtin_amdgcn_wmma_f32_16x16x4_f32(false, xf, false, wa,
                                                 (short)0, accA, false, false);
    accB = __builtin_amdgcn_wmma_f32_16x16x4_f32(false, xf, false, wb,
                                                 (short)0, accB, false, false);
  }

  const float bias = b1[h0 + r];   // column bias, same for all 8 rows
  float* aout = Amat + h0 + r;
  float* bout = Bmat + h0 + r;
  #pragma unroll
  for (int v = 0; v < 8; ++v) {
    const int row = i0 + v + (kq << 3);
    aout[row * H_] = accA[v] + bias;
    bout[row * H_] = accB[v];
  }
}

// ---------------------------------------------------------------------------
// Kernel 2 (dominant, ~4e8 lane-ops):
//   scores[i,j] = sum_h relu(A[i,h] + B[j,h]) * w2[h] + b2   (b1 pre-folded)
// 256-thread block -> 32x32 tile, 2x2 register blocking per thread (3 VALU
// ops per (h, output): add / max / fma). h chunked through DOUBLE-BUFFERED
// LDS via GLOBAL_LOAD_ASYNC_TO_LDS_B128 (ASYNCcnt), so chunk k+1 streams in
// while chunk k computes. Pitch 132 floats: 16B-aligned rows + 4-bank stagger.
// ---------------------------------------------------------------------------
__global__ void __launch_bounds__(256)
scores_kernel(const float* __restrict__ Amat, const float* __restrict__ Bmat,
              const float* __restrict__ W2,   const float* __restrict__ b2,
              float* __restrict__ out)
{
  constexpr int CH    = 128;
  constexpr int PITCH = CH + 4;            // 132 floats = 528 B (16B multiple)
  constexpr int NCH   = H_ / CH;           // 4 chunks
  __shared__ alignas(16) float sA[2][32][PITCH];
  __shared__ alignas(16) float sB[2][32][PITCH];
  __shared__ alignas(16) float sW[H_];

  const int tx  = threadIdx.x, ty = threadIdx.y;
  const int tid = ty * 16 + tx;
  const int i0  = blockIdx.y * 32;
  const int j0  = blockIdx.x * 32;
  const float b2v = b2[0];

  // w2 resident in LDS for the whole kernel (2 KB)
  sW[tid]       = W2[tid];
  sW[tid + 256] = W2[tid + 256];

  auto stage = [&](int chunk, int buf) {
    #pragma unroll
    for (int rep = 0; rep < 4; ++rep) {
      const int idx = tid + rep * 256;     // 1024 float4s per matrix chunk
      const int row = idx >> 5;            // 0..31
      const int c   = (idx & 31) << 2;     // 0..124
      const float* ga = Amat + (i0 + row) * H_ + chunk * CH + c;
      const float* gb = Bmat + (j0 + row) * H_ + chunk * CH + c;
#if USE_ASYNC_LDS
      __builtin_amdgcn_global_load_async_to_lds_b128(
          (gv4i*)ga, (lv4i*)&sA[buf][row][c], 0, 0);
      __builtin_amdgcn_global_load_async_to_lds_b128(
          (gv4i*)gb, (lv4i*)&sB[buf][row][c], 0, 0);
#else
      *reinterpret_cast<v4f*>(&sA[buf][row][c]) =
          *reinterpret_cast<const v4f*>(ga);
      *reinterpret_cast<v4f*>(&sB[buf][row][c]) =
          *reinterpret_cast<const v4f*>(gb);
#endif
    }
  };

  float acc00 = 0.f, acc01 = 0.f, acc10 = 0.f, acc11 = 0.f;

  stage(0, 0);                             // prologue: fill buffer 0
#if USE_ASYNC_LDS
  wait_async_lds();
#endif
  __syncthreads();

  for (int k = 0; k < NCH; ++k) {
    const int cur = k & 1;
    if (k + 1 < NCH) stage(k + 1, cur ^ 1);   // overlaps with compute below

    const v4f* a0  = reinterpret_cast<const v4f*>(&sA[cur][ty][0]);
    const v4f* a1  = reinterpret_cast<const v4f*>(&sA[cur][ty + 16][0]);
    const v4f* bb0 = reinterpret_cast<const v4f*>(&sB[cur][tx][0]);
    const v4f* bb1 = reinterpret_cast<const v4f*>(&sB[cur][tx + 16][0]);
    const v4f* wv  = reinterpret_cast<const v4f*>(&sW[k * CH]);

    #pragma unroll 8
    for (int q = 0; q < CH / 4; ++q) {
      const v4f A0 = a0[q], A1 = a1[q];
      const v4f B0 = bb0[q], B1 = bb1[q];
      const v4f Wq = wv[q];
      #pragma unroll
      for (int c = 0; c < 4; ++c) {
        acc00 += fmaxf(A0[c] + B0[c], 0.f) * Wq[c];
        acc01 += fmaxf(A0[c] + B1[c], 0.f) * Wq[c];
        acc10 += fmaxf(A1[c] + B0[c], 0.f) * Wq[c];
        acc11 += fmaxf(A1[c] + B1[c], 0.f) * Wq[c];
      }
    }

#if USE_ASYNC_LDS
    wait_async_lds();                      // only after compute: real overlap
#endif
    __syncthreads();
  }

  const int ia = i0 + ty, ib = i0 + ty + 16;
  const int ja = j0 + tx, jb = j0 + tx + 16;
  out[ia * N_ + ja] = (ia == ja) ? 0.f : acc00 + b2v;
  out[ia * N_ + jb] = (ia == jb) ? 0.f : acc01 + b2v;
  out[ib * N_ + ja] = (ib == ja) ? 0.f : acc10 + b2v;
  out[ib * N_ + jb] = (ib == jb) ? 0.f : acc11 + b2v;
}

extern "C" void kernel_launch(void* const* d_in, const int* in_sizes, int n_in,
                              void* d_out, int out_size, void* d_ws, size_t ws_size,
                              hipStream_t stream) {
  const float* X  = (const float*)d_in[0];   // (512, 1024)
  const float* W1 = (const float*)d_in[1];   // (512, 2048)
  const float* b1 = (const float*)d_in[2];   // (512,)
  const float* W2 = (const float*)d_in[3];   // (1, 512)
  const float* b2 = (const float*)d_in[4];   // (1,)
  float* out  = (float*)d_out;               // (512, 512)
  float* Amat = (float*)d_ws;                // 512*512 f32 (bias-folded A)
  float* Bmat = Amat + N_ * H_;              // 512*512 f32 (2 MB ws total)

  gemm_ab_kernel<<<128, 256, 0, stream>>>(X, W1, b1, Amat, Bmat);

  dim3 grid(16, 16), blk(16, 16);
  scores_kernel<<<grid, blk, 0, stream>>>(Amat, Bmat, W2, b2, out);
}